// LinearSpatialTemporal_82832739270908
// MI455X (gfx1250) — compile-verified
//
#include <hip/hip_runtime.h>

typedef __attribute__((ext_vector_type(2))) float v2f;
typedef __attribute__((ext_vector_type(8))) float v8f;
typedef __attribute__((ext_vector_type(2))) int   v2i;

#define PASTN 12
#define FUTN  12
#define Bn    64
#define Ln    256
#define Dn    512
#define Wn    244          // L - PAST
#define KC    32           // K chunk
#define NCH   (Dn / KC)    // 16 chunks
#define TW    16           // w values per block
#define MT    192          // rows per block = TW * FUTN
#define NTB   128          // output columns (e) per block
#define STT   34           // sT stride (even -> 8B-aligned float2, banks spread)
#define STBN  34           // sBn stride ([n][k] layout, even -> float2 B frags)

__launch_bounds__(256)
__global__ void lst_fused_kernel(const float* __restrict__ inp,
                                 const float* __restrict__ Wt,
                                 const float* __restrict__ bt,
                                 const float* __restrict__ Ws,
                                 const float* __restrict__ bs,
                                 float* __restrict__ out)
{
    __shared__ float sIn[(TW + PASTN - 1) * KC];       // 27 rows x 32 d (single buf)
    __shared__ float sT[MT * STT];                     // t tile: 192 x 32 (stride 34)
    __shared__ float sBn[2][NTB * STBN];               // W_s chunk: [n][k], double buf
    __shared__ float sWt[FUTN * PASTN];
    __shared__ float sBt[FUTN];

    const int tid   = threadIdx.x;
    const int lane  = tid & 31;
    const int wv    = tid >> 5;          // wave id 0..7
    const int n0    = blockIdx.x * NTB;  // e offset
    const int w0    = blockIdx.y * TW;   // w offset
    const int b     = blockIdx.z;

    if (tid < FUTN * PASTN) sWt[tid] = Wt[tid];
    if (tid < FUTN)         sBt[tid] = bt[tid];

    const int mg    = wv >> 1;            // 0..3 : 3 m-tiles each
    const int ng    = wv & 1;             // 0..1 : 4 n-tiles each
    const int mlane = lane & 15;
    const int khalf = (lane >> 4) << 1;   // 0 or 2 (A/B K placement per ISA layout)
    const int rhalf = (lane >> 4) << 3;   // 0 or 8 (C/D M placement per ISA layout)

    const v8f zero = {0.f, 0.f, 0.f, 0.f, 0.f, 0.f, 0.f, 0.f};
    v8f acc[3][4];
    #pragma unroll
    for (int i = 0; i < 3; ++i)
        #pragma unroll
        for (int j = 0; j < 4; ++j)
            acc[i][j] = zero;

    const int d  = tid & (KC - 1);        // 0..31
    const int rg = tid >> 5;              // row group 0..7 (24 t-rows each)

    // ---- async stage of one K-chunk: inp slice + W_s slice (b64 DMA to LDS) ----
    auto issue_stage = [&](int kc, int buf) {
        const int d0 = kc * KC;
        // sIn: 27 rows x 16 float2 = 432 b64 transfers
        #pragma unroll
        for (int it = 0; it < 2; ++it) {
            int idx2 = tid + 256 * it;
            if (idx2 < (TW + PASTN - 1) * (KC / 2)) {
                int rr  = idx2 >> 4;
                int dd2 = idx2 & 15;
                int grow = w0 + rr; if (grow > Ln - 1) grow = Ln - 1;
                const float* g = &inp[((size_t)b * Ln + grow) * Dn + d0 + dd2 * 2];
                float* l = &sIn[rr * KC + dd2 * 2];
                __builtin_amdgcn_global_load_async_to_lds_b64(
                    (v2i*)g, (v2i*)l, 0, 0);
            }
        }
        // sBn: 128 n-rows x 16 float2 = 2048 b64 transfers
        #pragma unroll
        for (int it = 0; it < 8; ++it) {
            int idx2 = tid + 256 * it;
            int k2 = idx2 & 15;
            int n  = idx2 >> 4;
            const float* g = &Ws[((size_t)(n0 + n)) * Dn + d0 + k2 * 2];
            float* l = &sBn[buf][n * STBN + k2 * 2];
            __builtin_amdgcn_global_load_async_to_lds_b64(
                (v2i*)g, (v2i*)l, 0, 0);
        }
    };

    issue_stage(0, 0);

    for (int kc = 0; kc < NCH; ++kc) {
        const int buf = kc & 1;

        __builtin_amdgcn_s_wait_asynccnt(0);   // chunk kc data landed in LDS
        __syncthreads();                       // visible to all waves; sT free

        // ---- temporal stage: t[r][d] = b_t[f] + sum_p Wt[f][p]*in[wl+p][d] ----
        #pragma unroll
        for (int j = 0; j < 24; ++j) {
            int f  = j % FUTN;
            int wl = rg * 2 + j / FUTN;
            float t = sBt[f];
            #pragma unroll
            for (int p = 0; p < PASTN; ++p)
                t += sWt[f * PASTN + p] * sIn[(wl + p) * KC + d];
            sT[(rg * 24 + j) * STT + d] = t;
        }
        __syncthreads();                       // sIn free, sT ready

        // overlap next chunk's DMA with this chunk's WMMA phase
        if (kc + 1 < NCH) issue_stage(kc + 1, buf ^ 1);

        // ---- spatial GEMM: 8 k-steps of 4, 12 WMMA tiles per wave ----
        #pragma unroll
        for (int ks = 0; ks < KC; ks += 4) {
            v2f a[3];
            #pragma unroll
            for (int i = 0; i < 3; ++i) {
                int row = (mg * 3 + i) * 16 + mlane;
                a[i] = *(const v2f*)&sT[row * STT + ks + khalf];
            }
            v2f bb[4];
            #pragma unroll
            for (int j = 0; j < 4; ++j) {
                int n_local = ng * 64 + j * 16 + mlane;
                bb[j] = *(const v2f*)&sBn[buf][n_local * STBN + ks + khalf];
            }
            #pragma unroll
            for (int i = 0; i < 3; ++i)
                #pragma unroll
                for (int j = 0; j < 4; ++j)
                    acc[i][j] = __builtin_amdgcn_wmma_f32_16x16x4_f32(
                        false, a[i], false, bb[j], (short)0, acc[i][j],
                        false, false);
        }
    }

    // ---- epilogue: + b_s, nontemporal streaming store ----
    float bsv[4];
    #pragma unroll
    for (int j = 0; j < 4; ++j)
        bsv[j] = bs[n0 + ng * 64 + j * 16 + mlane];

    #pragma unroll
    for (int i = 0; i < 3; ++i) {
        #pragma unroll
        for (int v = 0; v < 8; ++v) {
            int r  = (mg * 3 + i) * 16 + v + rhalf;   // local row in 0..191
            int wl = r / FUTN;
            int f  = r % FUTN;
            int w  = w0 + wl;
            if (w < Wn) {
                #pragma unroll
                for (int j = 0; j < 4; ++j) {
                    int e = n0 + ng * 64 + j * 16 + mlane;
                    size_t o = ((((size_t)b * Wn + w) * FUTN + f) * (size_t)Dn) + e;
                    __builtin_nontemporal_store(acc[i][j][v] + bsv[j], &out[o]);
                }
            }
        }
    }
}

extern "C" void kernel_launch(void* const* d_in, const int* in_sizes, int n_in,
                              void* d_out, int out_size, void* d_ws, size_t ws_size,
                              hipStream_t stream) {
    const float* inp = (const float*)d_in[0];
    const float* W_t = (const float*)d_in[1];
    const float* b_t = (const float*)d_in[2];
    const float* W_s = (const float*)d_in[3];
    const float* b_s = (const float*)d_in[4];
    float* out = (float*)d_out;
    (void)in_sizes; (void)n_in; (void)out_size; (void)d_ws; (void)ws_size;

    dim3 grid(Dn / NTB, (Wn + TW - 1) / TW, Bn);   // (4, 16, 64)
    dim3 block(256);
    hipLaunchKernelGGL(lst_fused_kernel, grid, block, 0, stream,
                       inp, W_t, b_t, W_s, b_s, out);
}